// SpectralEncoder_46213848105951
// MI455X (gfx1250) — compile-verified
//
#include <hip/hip_runtime.h>
#include <hip/hip_bf16.h>
#include <math.h>

typedef __attribute__((ext_vector_type(2))) float v2f;
typedef __attribute__((ext_vector_type(8))) float v8f;

#define B_DIM   32
#define T_DIM   4096
#define C_DIM   64
#define D_DIM   512
#define WIN     256
#define HOP     128
#define NW      31          // (4096-256)/128 + 1
#define NROWS   992         // B*NW  (= 62*16 exactly)
#define MROWS   100         // 50 bins * (re,im)
#define MPAD    128         // padded to 8 m-tiles of 16
#define NB      5
#define FEATW   (NB*C_DIM)  // 320

static __device__ __forceinline__ v8f wmma_f32(v2f a, v2f b, v8f c) {
    return __builtin_amdgcn_wmma_f32_16x16x4_f32(false, a, false, b, (short)0, c,
                                                 false, false);
}

// ---------------------------------------------------------------------------
// Kernel 0: trig table (MPAD x WIN), rows 2i = cos(2*pi*(i+1)*t/W),
//           rows 2i+1 = sin(...), rows 100..127 zero padding.
// ---------------------------------------------------------------------------
__global__ __launch_bounds__(256) void trig_fill(float* __restrict__ trig) {
    int idx = blockIdx.x * 256 + threadIdx.x;      // MPAD*WIN = 32768 entries
    if (idx >= MPAD * WIN) return;
    int m = idx >> 8;
    int t = idx & (WIN - 1);
    float v = 0.0f;
    if (m < MROWS) {
        int bin = (m >> 1) + 1;                    // bins 1..50
        float ang = (2.0f * 3.14159265358979323846f / (float)WIN) *
                    (float)bin * (float)t;
        v = (m & 1) ? sinf(ang) : cosf(ang);
    }
    trig[idx] = v;
}

// ---------------------------------------------------------------------------
// Kernel 1: per-window DFT via f32 WMMA + band-power reduction.
// One workgroup (8 waves) per (b, window). P = Trig(128x256) @ Win(256x64),
// tiled 16x16 with K-steps of 4; P staged in LDS; then 320 band features.
// ---------------------------------------------------------------------------
__global__ __launch_bounds__(256) void dft_band(const float* __restrict__ x,
                                                const float* __restrict__ trig,
                                                float* __restrict__ feat) {
    __shared__ float P[MPAD * C_DIM];              // 32 KB
    const int wg = blockIdx.x;                     // 0..991
    const int b = wg / NW;
    const int n = wg % NW;
    const float* win = x + ((size_t)b * T_DIM + (size_t)n * HOP) * C_DIM;

    const int tid  = threadIdx.x;
    const int wave = tid >> 5;
    const int lane = tid & 31;
    const int lm = lane & 15;
    const int hi = lane >> 4;                      // lane half: 0 or 1

    // 8 m-tiles x 4 n-tiles = 32 tiles; 8 waves -> 4 tiles each
    for (int i = 0; i < 4; ++i) {
        const int tile = wave * 4 + i;
        const int m0 = (tile >> 2) * 16;
        const int n0 = (tile & 3) * 16;
        v8f acc = {};
        const float* arow = trig + (size_t)(m0 + lm) * WIN + 2 * hi;
        const float* bcol = win + n0 + lm;
        for (int k = 0; k < WIN; k += 4) {
            v2f a;  a.x = arow[k];  a.y = arow[k + 1];
            v2f bb;
            bb.x = bcol[(size_t)(k + 2 * hi) * C_DIM];
            bb.y = bcol[(size_t)(k + 2 * hi + 1) * C_DIM];
            acc = wmma_f32(a, bb, acc);
        }
        const int row0 = m0 + hi * 8;
        const int col  = n0 + lm;
        #pragma unroll
        for (int r = 0; r < 8; ++r)
            P[(row0 + r) * C_DIM + col] = acc[r];
    }
    __syncthreads();

    // band reduction: bin index i2 = bin-1; inclusive ranges from the masks
    const int   lo[NB]  = {0, 3, 7, 12, 29};
    const int   hb[NB]  = {3, 7, 12, 29, 49};
    const float inv[NB] = {1.0f/4.0f, 1.0f/5.0f, 1.0f/6.0f, 1.0f/18.0f, 1.0f/21.0f};
    for (int idx = tid; idx < FEATW; idx += 256) {
        const int band = idx >> 6;
        const int c    = idx & (C_DIM - 1);
        float s = 0.0f;
        for (int i2 = lo[band]; i2 <= hb[band]; ++i2) {
            const float re = P[(2 * i2) * C_DIM + c];
            const float im = P[(2 * i2 + 1) * C_DIM + c];
            s += re * re + im * im;
        }
        feat[(size_t)wg * FEATW + idx] = s * inv[band];
    }
}

// ---------------------------------------------------------------------------
// Kernel 2: fused 2-layer MLP. One workgroup per 16-row M-tile (62 WGs).
// GEMM1 (K=320,N=256) -> +b1, relu -> LDS H(16x256) -> GEMM2 (K=256,N=512).
// ---------------------------------------------------------------------------
__global__ __launch_bounds__(256) void mlp(const float* __restrict__ feat,
                                           const float* __restrict__ W1,
                                           const float* __restrict__ b1,
                                           const float* __restrict__ W2,
                                           const float* __restrict__ b2,
                                           float* __restrict__ outF) {
    __shared__ float H[16 * 256];                  // 16 KB
    const int m0g  = blockIdx.x * 16;
    const int tid  = threadIdx.x;
    const int wave = tid >> 5;
    const int lane = tid & 31;
    const int lm = lane & 15;
    const int hi = lane >> 4;

    // --- layer 1: N=256 -> 16 n-tiles, 2 per wave ---
    for (int i = 0; i < 2; ++i) {
        const int n0 = (wave * 2 + i) * 16;
        v8f acc = {};
        const float* arow = feat + (size_t)(m0g + lm) * FEATW + 2 * hi;
        for (int k = 0; k < FEATW; k += 4) {
            v2f a;  a.x = arow[k];  a.y = arow[k + 1];
            v2f bb;
            bb.x = W1[(size_t)(k + 2 * hi) * 256 + n0 + lm];
            bb.y = W1[(size_t)(k + 2 * hi + 1) * 256 + n0 + lm];
            acc = wmma_f32(a, bb, acc);
        }
        const int col = n0 + lm;
        const float bias = b1[col];
        #pragma unroll
        for (int r = 0; r < 8; ++r) {
            float v = acc[r] + bias;
            H[(hi * 8 + r) * 256 + col] = v > 0.0f ? v : 0.0f;
        }
    }
    __syncthreads();

    // --- layer 2: N=512 -> 32 n-tiles, 4 per wave ---
    for (int i = 0; i < 4; ++i) {
        const int n0 = (wave * 4 + i) * 16;
        v8f acc = {};
        const float* arow = &H[lm * 256 + 2 * hi];
        for (int k = 0; k < 256; k += 4) {
            v2f a;  a.x = arow[k];  a.y = arow[k + 1];
            v2f bb;
            bb.x = W2[(size_t)(k + 2 * hi) * D_DIM + n0 + lm];
            bb.y = W2[(size_t)(k + 2 * hi + 1) * D_DIM + n0 + lm];
            acc = wmma_f32(a, bb, acc);
        }
        const int col = n0 + lm;
        const float bias = b2[col];
        #pragma unroll
        for (int r = 0; r < 8; ++r)
            outF[(size_t)(m0g + hi * 8 + r) * D_DIM + col] = acc[r] + bias;
    }
}

// ---------------------------------------------------------------------------
// Kernel 3: linear interpolation nW=31 -> T=4096 (align_corners=False).
// Pure streaming float4 writes (256 MB) — the pipeline's bandwidth floor.
// ---------------------------------------------------------------------------
__global__ __launch_bounds__(256) void interp(const float* __restrict__ outF,
                                              float* __restrict__ out) {
    const size_t gid = (size_t)blockIdx.x * 256 + threadIdx.x; // float4 units
    const int d4 = (int)(gid & 127);                           // D/4 = 128
    const int t  = (int)((gid >> 7) & (T_DIM - 1));
    const int b  = (int)(gid >> 19);

    float pos = ((float)t + 0.5f) * ((float)NW / (float)T_DIM) - 0.5f;
    pos = fminf(fmaxf(pos, 0.0f), (float)(NW - 1));
    const float f0 = floorf(pos);
    const int i0 = (int)f0;
    const int i1 = (i0 + 1 < NW) ? (i0 + 1) : (NW - 1);
    const float w = pos - f0;

    const float4* r0 = (const float4*)(outF + (size_t)(b * NW + i0) * D_DIM) + d4;
    const float4* r1 = (const float4*)(outF + (size_t)(b * NW + i1) * D_DIM) + d4;
    const float4 v0 = *r0;
    const float4 v1 = *r1;
    float4 o;
    o.x = v0.x + (v1.x - v0.x) * w;
    o.y = v0.y + (v1.y - v0.y) * w;
    o.z = v0.z + (v1.z - v0.z) * w;
    o.w = v0.w + (v1.w - v0.w) * w;
    ((float4*)out)[gid] = o;
}

// ---------------------------------------------------------------------------
extern "C" void kernel_launch(void* const* d_in, const int* in_sizes, int n_in,
                              void* d_out, int out_size, void* d_ws, size_t ws_size,
                              hipStream_t stream) {
    const float* x  = (const float*)d_in[0];
    const float* W1 = (const float*)d_in[1];
    const float* b1 = (const float*)d_in[2];
    const float* W2 = (const float*)d_in[3];
    const float* b2 = (const float*)d_in[4];
    float* out = (float*)d_out;

    // workspace layout (bytes): trig | feat | outF
    char* ws = (char*)d_ws;
    float* trig = (float*)(ws);                                   // 128*256*4 = 128 KB
    float* feat = (float*)(ws + (size_t)MPAD * WIN * 4);          // 992*320*4
    float* outF = (float*)(ws + (size_t)MPAD * WIN * 4
                              + (size_t)NROWS * FEATW * 4);       // 992*512*4

    trig_fill<<<(MPAD * WIN + 255) / 256, 256, 0, stream>>>(trig);
    dft_band<<<NROWS, 256, 0, stream>>>(x, trig, feat);
    mlp<<<NROWS / 16, 256, 0, stream>>>(feat, W1, b1, W2, b2, outF);

    const size_t nvec4 = (size_t)B_DIM * T_DIM * (D_DIM / 4);     // 16,777,216
    interp<<<(unsigned)(nvec4 / 256), 256, 0, stream>>>(outF, out);
}